// GCN_89773406421550
// MI455X (gfx1250) — compile-verified
//
#include <hip/hip_runtime.h>
#include <hip/hip_bf16.h>
#include <math.h>

typedef __attribute__((ext_vector_type(2))) float v2f;
typedef __attribute__((ext_vector_type(8))) float v8f;

#define NG    256      // graphs
#define HID   256      // hidden channels
#define ADDF  7        // extra graph features
#define KIN   263      // HID + ADDF
#define KPAD  264      // padded to multiple of 4 for 16x16x4 wmma
#define MLP1  128
#define GN_EPS 1e-5f

__device__ __forceinline__ float gelu_exact(float v) {
    return 0.5f * v * (1.0f + erff(v * 0.7071067811865475f));
}
__device__ __forceinline__ float sigmoid_(float v) {
    return 1.0f / (1.0f + expf(-v));
}

// ---------- init ----------
__global__ void k_init_nodes(float* __restrict__ deg, float* __restrict__ s, int n) {
    int i = blockIdx.x * blockDim.x + threadIdx.x;
    if (i < n) { deg[i] = 1.0f; s[i] = 0.0f; }   // deg starts at 1 (self loop)
}

__global__ void k_init_stats(float* __restrict__ S, float* __restrict__ counts,
                             float* __restrict__ T) {
    int i = blockIdx.x * blockDim.x + threadIdx.x;
    if (i < 2 * HID) S[i] = 0.0f;
    if (i < NG)      counts[i] = 0.0f;
    if (i < NG * HID) T[i] = 0.0f;
}

__global__ void k_prep_w(const float* __restrict__ W1, const float* __restrict__ W2,
                         float* __restrict__ W1p, float* __restrict__ W2p) {
    int i = blockIdx.x * blockDim.x + threadIdx.x;
    if (i < KPAD * MLP1) {
        int r = i / MLP1, c = i % MLP1;
        W1p[i] = (r < KIN) ? W1[r * MLP1 + c] : 0.0f;
    }
    if (i < MLP1 * 16) {
        int r = i / 16, c = i % 16;
        W2p[i] = (c < 2) ? W2[r * 2 + c] : 0.0f;
    }
}

// ---------- graph scatter phase ----------
__global__ void k_deg(const int* __restrict__ dst, float* __restrict__ deg, int E) {
    int e = blockIdx.x * blockDim.x + threadIdx.x;
    if (e < E) atomicAdd(&deg[dst[e]], 1.0f);
}

__global__ void k_dinv_counts(const float* __restrict__ deg, float* __restrict__ dinv,
                              const int* __restrict__ batch, float* __restrict__ counts, int n) {
    int i = blockIdx.x * blockDim.x + threadIdx.x;
    if (i < n) {
        dinv[i] = rsqrtf(deg[i]);          // deg >= 1 always (self loop)
        atomicAdd(&counts[batch[i]], 1.0f);
    }
}

__global__ void k_scatter(const int* __restrict__ src, const int* __restrict__ dst,
                          const float* __restrict__ x, const float* __restrict__ dinv,
                          float* __restrict__ s, int E) {
    int e = blockIdx.x * blockDim.x + threadIdx.x;
    if (e < E) {
        int a = src[e], b = dst[e];
        atomicAdd(&s[b], dinv[a] * dinv[b] * x[a]);
    }
}

// ---------- per-(node,channel) GELU + reductions ----------
// block = 256 threads: thread == channel; strip of 256 nodes broadcast via LDS.
// batch is sorted, so T-atomics flush only on graph-id change within the strip.
__global__ void k_node(const float* __restrict__ s, const float* __restrict__ dinv,
                       const float* __restrict__ x, const int* __restrict__ batch,
                       const float* __restrict__ Wc, const float* __restrict__ bc,
                       float* __restrict__ S, float* __restrict__ T, int n) {
    __shared__ float s_sh[256];
    __shared__ int   b_sh[256];
    int base = blockIdx.x * 256;
    int cnt  = n - base; if (cnt > 256) cnt = 256;
    int tid  = threadIdx.x;
    if (tid < cnt) {
        int i = base + tid;
        float di = dinv[i];
        s_sh[tid] = s[i] + di * di * x[i];   // add self-loop term
        b_sh[tid] = batch[i];
    }
    __syncthreads();
    if (cnt <= 0) return;
    float w = Wc[tid], b = bc[tid];
    float a1 = 0.0f, a2 = 0.0f, aT = 0.0f;
    int curg = b_sh[0];
    for (int k = 0; k < cnt; ++k) {
        int g = b_sh[k];
        if (g != curg) {
            atomicAdd(&T[curg * HID + tid], aT);
            aT = 0.0f; curg = g;
        }
        float v = gelu_exact(fmaf(s_sh[k], w, b));
        a1 += v; a2 += v * v; aT += v;
    }
    atomicAdd(&T[curg * HID + tid], aT);
    atomicAdd(&S[tid], a1);
    atomicAdd(&S[HID + tid], a2);
}

// ---------- BN + pool + concat -> G [256 x 264] ----------
__global__ void k_build_G(const float* __restrict__ S, const float* __restrict__ counts,
                          const float* __restrict__ T, const float* __restrict__ gamma,
                          const float* __restrict__ beta, const float* __restrict__ yfeat,
                          float* __restrict__ G, float invN) {
    int idx = blockIdx.x * blockDim.x + threadIdx.x;
    if (idx >= NG * KPAD) return;
    int g = idx / KPAD, c = idx % KPAD;
    float val;
    if (c < HID) {
        float mean  = S[c] * invN;
        float var   = S[HID + c] * invN - mean * mean;
        float scale = rsqrtf(var + GN_EPS) * gamma[c];
        float cg    = fmaxf(counts[g], 1.0f);
        val = (T[g * HID + c] / cg - mean) * scale + beta[c];
    } else if (c < KIN) {
        val = yfeat[g * ADDF + (c - HID)];
    } else {
        val = 0.0f;
    }
    G[idx] = val;
}

// ---------- GEMM1: [256x264] @ [264x128] + b1, gelu -> H [256x128] ----------
// f32 WMMA 16x16x4; 128 waves, one 16x16 tile each.
__global__ void k_gemm1(const float* __restrict__ G, const float* __restrict__ W1p,
                        const float* __restrict__ b1, float* __restrict__ H) {
    int lane = threadIdx.x & 31;
    int wave = blockIdx.x * (blockDim.x >> 5) + (threadIdx.x >> 5);
    int tM = wave >> 3;          // 0..15
    int tN = wave & 7;           // 0..7
    int half = lane >> 4;        // lane half selects K pair / M+8
    int l = lane & 15;
    int arow = tM * 16 + l;
    int bcol = tN * 16 + l;
    v8f acc = {};
    for (int k0 = 0; k0 < KPAD; k0 += 4) {
        int ka = k0 + half * 2;
        v2f a, b;
        a.x = G[arow * KPAD + ka];
        a.y = G[arow * KPAD + ka + 1];
        b.x = W1p[ka * MLP1 + bcol];
        b.y = W1p[(ka + 1) * MLP1 + bcol];
        acc = __builtin_amdgcn_wmma_f32_16x16x4_f32(false, a, false, b,
                                                    (short)0, acc, false, false);
    }
    float bias = b1[bcol];
#pragma unroll
    for (int r = 0; r < 8; ++r) {
        int m = tM * 16 + r + half * 8;
        H[m * MLP1 + bcol] = gelu_exact(acc[r] + bias);
    }
}

// ---------- GEMM2: [256x128] @ [128x16(pad of 2)] + b2, sigmoid -> out [256x2] ----------
__global__ void k_gemm2(const float* __restrict__ H, const float* __restrict__ W2p,
                        const float* __restrict__ b2, float* __restrict__ out) {
    int lane = threadIdx.x & 31;
    int wave = threadIdx.x >> 5;     // blockDim = 512 -> 16 waves
    int tM = wave;
    int half = lane >> 4;
    int l = lane & 15;
    int arow = tM * 16 + l;
    v8f acc = {};
    for (int k0 = 0; k0 < MLP1; k0 += 4) {
        int ka = k0 + half * 2;
        v2f a, b;
        a.x = H[arow * MLP1 + ka];
        a.y = H[arow * MLP1 + ka + 1];
        b.x = W2p[ka * 16 + l];
        b.y = W2p[(ka + 1) * 16 + l];
        acc = __builtin_amdgcn_wmma_f32_16x16x4_f32(false, a, false, b,
                                                    (short)0, acc, false, false);
    }
    // EXEC was all-ones for every WMMA above; mask only the store.
    if (l < 2) {
        float bias = b2[l];
#pragma unroll
        for (int r = 0; r < 8; ++r) {
            int m = tM * 16 + r + half * 8;
            out[m * 2 + l] = sigmoid_(acc[r] + bias);
        }
    }
}

extern "C" void kernel_launch(void* const* d_in, const int* in_sizes, int n_in,
                              void* d_out, int out_size, void* d_ws, size_t ws_size,
                              hipStream_t stream) {
    const float* x     = (const float*)d_in[0];
    const int*   ei    = (const int*)d_in[1];
    const int*   batch = (const int*)d_in[2];
    const float* yfeat = (const float*)d_in[3];
    const float* Wc    = (const float*)d_in[4];
    const float* bc    = (const float*)d_in[5];
    const float* gamma = (const float*)d_in[6];
    const float* beta  = (const float*)d_in[7];
    const float* W1    = (const float*)d_in[8];
    const float* b1    = (const float*)d_in[9];
    const float* W2    = (const float*)d_in[10];
    const float* b2    = (const float*)d_in[11];

    const int N = in_sizes[0];
    const int E = in_sizes[1] / 2;
    const int* src = ei;
    const int* dst = ei + E;

    // workspace layout (floats)
    float* w      = (float*)d_ws;
    float* deg    = w;                    // N
    float* dinv   = deg    + N;           // N
    float* s      = dinv   + N;           // N
    float* S      = s      + N;           // 2*HID (S1,S2)
    float* counts = S      + 2 * HID;     // NG
    float* T      = counts + NG;          // NG*HID
    float* G      = T      + NG * HID;    // NG*KPAD
    float* W1p    = G      + NG * KPAD;   // KPAD*MLP1
    float* W2p    = W1p    + KPAD * MLP1; // MLP1*16
    float* H      = W2p    + MLP1 * 16;   // NG*MLP1

    const int TB = 256;
    k_init_nodes<<<(N + TB - 1) / TB, TB, 0, stream>>>(deg, s, N);
    k_init_stats<<<(NG * HID + TB - 1) / TB, TB, 0, stream>>>(S, counts, T);
    k_prep_w<<<(KPAD * MLP1 + TB - 1) / TB, TB, 0, stream>>>(W1, W2, W1p, W2p);
    k_deg<<<(E + TB - 1) / TB, TB, 0, stream>>>(dst, deg, E);
    k_dinv_counts<<<(N + TB - 1) / TB, TB, 0, stream>>>(deg, dinv, batch, counts, N);
    k_scatter<<<(E + TB - 1) / TB, TB, 0, stream>>>(src, dst, x, dinv, s, E);
    k_node<<<(N + TB - 1) / TB, TB, 0, stream>>>(s, dinv, x, batch, Wc, bc, S, T, N);
    k_build_G<<<(NG * KPAD + TB - 1) / TB, TB, 0, stream>>>(S, counts, T, gamma, beta,
                                                            yfeat, G, 1.0f / (float)N);
    k_gemm1<<<16, 256, 0, stream>>>(G, W1p, b1, H);   // 128 waves, 16x8 tiles
    k_gemm2<<<1, 512, 0, stream>>>(H, W2p, b2, (float*)d_out); // 16 waves
}